// TFCalibrationKD_37830071943325
// MI455X (gfx1250) — compile-verified
//
#include <hip/hip_runtime.h>
#include <cmath>

typedef __attribute__((ext_vector_type(2))) float v2f;
typedef __attribute__((ext_vector_type(8))) float v8f;

#define B_DIM 8
#define C_DIM 64
#define T_DIM 192
#define F_DIM 128
#define EPS_F 1e-8f
#define LN_EPS_F 1e-5f
#define NSPLIT 4

// ---------------------------------------------------------------------------
// Gram over time: G[b](192x192) = sum_c M_c M_c^T, M_c = feat[b,c] (192x128)
// Block = 256 threads (8 waves), computes a 64x64 output tile; split-K over c.
// ---------------------------------------------------------------------------
__global__ __launch_bounds__(256) void gram_time_kernel(const float* __restrict__ feat,
                                                        float* __restrict__ Gpart) {
  __shared__ alignas(16) float ldsA[64 * 68];   // stride 68 -> conflict-free b64 reads
  __shared__ alignas(16) float ldsB[64 * 68];
  const int tid  = threadIdx.x;
  const int lane = tid & 31;
  const int wave = tid >> 5;
  const int wr   = wave >> 2;          // 0..1
  const int wc   = wave & 3;           // 0..3
  const int lm   = lane & 15;
  const int koff = (lane >> 4) << 1;   // 0 or 2
  const int b     = blockIdx.z & 7;
  const int split = blockIdx.z >> 3;
  const int m0 = blockIdx.y * 64;
  const int n0 = blockIdx.x * 64;

  v8f acc0 = {}; v8f acc1 = {};
  const float* base = feat + (size_t)b * (C_DIM * T_DIM * F_DIM);

  for (int c = split * (C_DIM / NSPLIT); c < (split + 1) * (C_DIM / NSPLIT); ++c) {
    const float* slab = base + (size_t)c * (T_DIM * F_DIM);
    for (int fc = 0; fc < 2; ++fc) {               // two 64-wide K chunks of the 128 freqs
      // vectorized coalesced LDS fill: b128 loads/stores
      for (int i = 0; i < 4; ++i) {
        int idx = i * 256 + tid;                   // 0..1023 float4s
        int r  = idx >> 4;                         // row 0..63
        int f4 = (idx & 15) << 2;                  // 0,4,...,60
        *(float4*)&ldsA[r * 68 + f4] = *(const float4*)&slab[(m0 + r) * F_DIM + fc * 64 + f4];
        *(float4*)&ldsB[r * 68 + f4] = *(const float4*)&slab[(n0 + r) * F_DIM + fc * 64 + f4];
      }
      __syncthreads();
      const float* pa0 = &ldsA[(wr * 16 + lm) * 68 + koff];
      const float* pa1 = &ldsA[((wr + 2) * 16 + lm) * 68 + koff];
      const float* pb  = &ldsB[(wc * 16 + lm) * 68 + koff];
      for (int k = 0; k < 64; k += 4) {
        v2f a0 = *(const v2f*)(pa0 + k);
        v2f a1 = *(const v2f*)(pa1 + k);
        v2f bb = *(const v2f*)(pb + k);
        acc0 = __builtin_amdgcn_wmma_f32_16x16x4_f32(false, a0, false, bb, (short)0, acc0, false, false);
        acc1 = __builtin_amdgcn_wmma_f32_16x16x4_f32(false, a1, false, bb, (short)0, acc1, false, false);
      }
      __syncthreads();
    }
  }
  float* Gb = Gpart + (size_t)split * (B_DIM * T_DIM * T_DIM) + (size_t)b * (T_DIM * T_DIM);
  const int col = n0 + wc * 16 + lm;
  const int half8 = (lane >> 4) << 3;
  for (int v = 0; v < 8; ++v) {
    int r0 = m0 + wr * 16 + v + half8;
    int r1 = m0 + (wr + 2) * 16 + v + half8;
    Gb[(size_t)r0 * T_DIM + col] = acc0[v];
    Gb[(size_t)r1 * T_DIM + col] = acc1[v];
  }
}

// ---------------------------------------------------------------------------
// Gram over freq: G[b](128x128) = Y^T Y, Y = feat[b] viewed (12288,128) row-major.
// LDS tile [k][f] (stride 72), split-K over 64-row k-chunks.
// ---------------------------------------------------------------------------
__global__ __launch_bounds__(256) void gram_freq_kernel(const float* __restrict__ feat,
                                                        float* __restrict__ Gpart) {
  __shared__ alignas(16) float ldsA[64 * 72];
  __shared__ alignas(16) float ldsB[64 * 72];
  const int tid  = threadIdx.x;
  const int lane = tid & 31;
  const int wave = tid >> 5;
  const int wr   = wave >> 2;
  const int wc   = wave & 3;
  const int lm   = lane & 15;
  const int koff = (lane >> 4) << 1;
  const int b     = blockIdx.z & 7;
  const int split = blockIdx.z >> 3;
  const int m0 = blockIdx.y * 64;
  const int n0 = blockIdx.x * 64;

  v8f acc0 = {}; v8f acc1 = {};
  const float* Y = feat + (size_t)b * (C_DIM * T_DIM * F_DIM);
  const int CHUNKS_PER_SPLIT = (C_DIM * T_DIM / 64) / NSPLIT;   // 48

  for (int ch = split * CHUNKS_PER_SPLIT; ch < (split + 1) * CHUNKS_PER_SPLIT; ++ch) {
    const int k0 = ch * 64;
    for (int i = 0; i < 4; ++i) {
      int idx = i * 256 + tid;
      int r  = idx >> 4;
      int f4 = (idx & 15) << 2;
      *(float4*)&ldsA[r * 72 + f4] = *(const float4*)&Y[(size_t)(k0 + r) * F_DIM + m0 + f4];
      *(float4*)&ldsB[r * 72 + f4] = *(const float4*)&Y[(size_t)(k0 + r) * F_DIM + n0 + f4];
    }
    __syncthreads();
    for (int k = 0; k < 64; k += 4) {
      int ra = (k + koff) * 72;
      int rb = (k + koff + 1) * 72;
      v2f a0, a1, bb;
      a0.x = ldsA[ra + wr * 16 + lm];        a0.y = ldsA[rb + wr * 16 + lm];
      a1.x = ldsA[ra + (wr + 2) * 16 + lm];  a1.y = ldsA[rb + (wr + 2) * 16 + lm];
      bb.x = ldsB[ra + wc * 16 + lm];        bb.y = ldsB[rb + wc * 16 + lm];
      acc0 = __builtin_amdgcn_wmma_f32_16x16x4_f32(false, a0, false, bb, (short)0, acc0, false, false);
      acc1 = __builtin_amdgcn_wmma_f32_16x16x4_f32(false, a1, false, bb, (short)0, acc1, false, false);
    }
    __syncthreads();
  }
  float* Gb = Gpart + (size_t)split * (B_DIM * F_DIM * F_DIM) + (size_t)b * (F_DIM * F_DIM);
  const int col = n0 + wc * 16 + lm;
  const int half8 = (lane >> 4) << 3;
  for (int v = 0; v < 8; ++v) {
    int r0 = m0 + wr * 16 + v + half8;
    int r1 = m0 + (wr + 2) * 16 + v + half8;
    Gb[(size_t)r0 * F_DIM + col] = acc0[v];
    Gb[(size_t)r1 * F_DIM + col] = acc1[v];
  }
}

// --------------------------- small support kernels -------------------------

__global__ void reduce4_kernel(const float* __restrict__ part, float* __restrict__ G, int n) {
  int i = blockIdx.x * blockDim.x + threadIdx.x;
  if (i < n) G[i] = part[i] + part[(size_t)n + i] + part[2 * (size_t)n + i] + part[3 * (size_t)n + i];
}

__global__ void norms_kernel(const float* __restrict__ G, float* __restrict__ norms, int L) {
  int i = blockIdx.x * blockDim.x + threadIdx.x;
  if (i < B_DIM * L) {
    int b = i / L, l = i % L;
    float d = G[((size_t)(b * L + l)) * L + l];
    norms[i] = fmaxf(sqrtf(fmaxf(d, 0.f)), EPS_F);
  }
}

// G -> P = 0.5*(cos+1) in place; also row means of P
__global__ void p_rowmean_kernel(float* __restrict__ G, const float* __restrict__ norms,
                                 float* __restrict__ rowmean, int L) {
  __shared__ float red[256];
  const int row = blockIdx.x;                 // b*L + l
  const int b = row / L;
  const int tid = threadIdx.x;
  const float nl = norms[row];
  float* Grow = G + (size_t)row * L;
  float s = 0.f;
  for (int j = tid; j < L; j += 256) {
    float p = 0.5f * (Grow[j] / (nl * norms[b * L + j]) + 1.0f);
    Grow[j] = p; s += p;
  }
  red[tid] = s; __syncthreads();
  for (int off = 128; off > 0; off >>= 1) { if (tid < off) red[tid] += red[tid + off]; __syncthreads(); }
  if (tid == 0) rowmean[row] = red[0] / (float)L;
}

// Linear(1,4) + ReLU + LayerNorm(4)
__global__ void mlp_kernel(const float* __restrict__ rm, const float* __restrict__ W,
                           const float* __restrict__ bias, const float* __restrict__ g,
                           const float* __restrict__ beta, float* __restrict__ out, int n) {
  int i = blockIdx.x * blockDim.x + threadIdx.x;
  if (i >= n) return;
  float x = rm[i];
  float h[4]; float mu = 0.f;
  for (int j = 0; j < 4; ++j) { h[j] = fmaxf(W[j] * x + bias[j], 0.f); mu += h[j]; }
  mu *= 0.25f;
  float var = 0.f;
  for (int j = 0; j < 4; ++j) { float d = h[j] - mu; var += d * d; }
  var *= 0.25f;
  float inv = rsqrtf(var + LN_EPS_F);
  for (int j = 0; j < 4; ++j) out[(size_t)i * 4 + j] = g[j] * (h[j] - mu) * inv + beta[j];
}

// alpha[b,s,:] = softmax_t( q[b,s]·k[b,t] / SCALE )
__global__ void alpha_kernel(const float* __restrict__ q, const float* __restrict__ k,
                             float* __restrict__ alpha, int L, float inv_scale) {
  __shared__ float red[256];
  const int row = blockIdx.x;               // b*L + s
  const int b = row / L;
  const int tid = threadIdx.x;
  const float* qs = q + (size_t)row * 4;
  const float q0 = qs[0], q1 = qs[1], q2 = qs[2], q3 = qs[3];
  float* arow = alpha + (size_t)row * L;
  float mx = -3.0e38f;
  for (int t = tid; t < L; t += 256) {
    const float* kt = k + (size_t)(b * L + t) * 4;
    float v = (q0 * kt[0] + q1 * kt[1] + q2 * kt[2] + q3 * kt[3]) * inv_scale;
    arow[t] = v; mx = fmaxf(mx, v);
  }
  red[tid] = mx; __syncthreads();
  for (int off = 128; off > 0; off >>= 1) { if (tid < off) red[tid] = fmaxf(red[tid], red[tid + off]); __syncthreads(); }
  mx = red[0]; __syncthreads();
  float s = 0.f;
  for (int t = tid; t < L; t += 256) { float e = expf(arow[t] - mx); arow[t] = e; s += e; }
  red[tid] = s; __syncthreads();
  for (int off = 128; off > 0; off >>= 1) { if (tid < off) red[tid] += red[tid + off]; __syncthreads(); }
  float inv = 1.0f / red[0];
  for (int t = tid; t < L; t += 256) arow[t] *= inv;
}

// s-side: P -> log(softmax(P)+EPS) in place
__global__ void softmax_log_kernel(float* __restrict__ P, int L) {
  __shared__ float red[256];
  float* pr = P + (size_t)blockIdx.x * L;
  const int tid = threadIdx.x;
  float mx = -3.0e38f;
  for (int j = tid; j < L; j += 256) mx = fmaxf(mx, pr[j]);
  red[tid] = mx; __syncthreads();
  for (int off = 128; off > 0; off >>= 1) { if (tid < off) red[tid] = fmaxf(red[tid], red[tid + off]); __syncthreads(); }
  mx = red[0]; __syncthreads();
  float s = 0.f;
  for (int j = tid; j < L; j += 256) { float e = expf(pr[j] - mx); pr[j] = e; s += e; }
  red[tid] = s; __syncthreads();
  for (int off = 128; off > 0; off >>= 1) { if (tid < off) red[tid] += red[tid + off]; __syncthreads(); }
  float inv = 1.0f / red[0];
  for (int j = tid; j < L; j += 256) pr[j] = logf(pr[j] * inv + EPS_F);
}

// t-side: P -> softmax(P)+EPS in place; ent[row] = sum p*log(p)
__global__ void softmax_ent_kernel(float* __restrict__ P, float* __restrict__ ent, int L) {
  __shared__ float red[256];
  const int row = blockIdx.x;
  float* pr = P + (size_t)row * L;
  const int tid = threadIdx.x;
  float mx = -3.0e38f;
  for (int j = tid; j < L; j += 256) mx = fmaxf(mx, pr[j]);
  red[tid] = mx; __syncthreads();
  for (int off = 128; off > 0; off >>= 1) { if (tid < off) red[tid] = fmaxf(red[tid], red[tid + off]); __syncthreads(); }
  mx = red[0]; __syncthreads();
  float s = 0.f;
  for (int j = tid; j < L; j += 256) { float e = expf(pr[j] - mx); pr[j] = e; s += e; }
  red[tid] = s; __syncthreads();
  for (int off = 128; off > 0; off >>= 1) { if (tid < off) red[tid] += red[tid + off]; __syncthreads(); }
  float inv = 1.0f / red[0]; __syncthreads();
  float loc = 0.f;
  for (int j = tid; j < L; j += 256) { float p = pr[j] * inv + EPS_F; pr[j] = p; loc += p * logf(p); }
  red[tid] = loc; __syncthreads();
  for (int off = 128; off > 0; off >>= 1) { if (tid < off) red[tid] += red[tid + off]; __syncthreads(); }
  if (tid == 0) ent[row] = red[0];
}

// partial[b*L+s] = sum_t alpha[b,s,t] * (ent[b,t] - sum_k Pt[b,t,k]*logPs[b,s,k])
__global__ void kl_loss_kernel(const float* __restrict__ Pt, const float* __restrict__ logPs,
                               const float* __restrict__ alpha, const float* __restrict__ ent,
                               float* __restrict__ partial, int L) {
  __shared__ float ldsLog[192];
  __shared__ float red[256];
  const int row = blockIdx.x;               // b*L + s
  const int b = row / L;
  const int tid = threadIdx.x;
  const float* ls = logPs + (size_t)row * L;
  for (int kk = tid; kk < L; kk += 256) ldsLog[kk] = ls[kk];
  __syncthreads();
  float acc = 0.f;
  for (int t = tid; t < L; t += 256) {
    const float* pt = Pt + (size_t)(b * L + t) * L;
    float cross = 0.f;
    for (int kk = 0; kk < L; ++kk) cross += pt[kk] * ldsLog[kk];
    acc += alpha[(size_t)row * L + t] * (ent[b * L + t] - cross);
  }
  red[tid] = acc; __syncthreads();
  for (int off = 128; off > 0; off >>= 1) { if (tid < off) red[tid] += red[tid + off]; __syncthreads(); }
  if (tid == 0) partial[row] = red[0];
}

__global__ void finalize_kernel(const float* __restrict__ pT, int nT,
                                const float* __restrict__ pF, int nF, float* __restrict__ out) {
  __shared__ float red[256];
  const int tid = threadIdx.x;
  float s = 0.f;
  for (int i = tid; i < nT; i += 256) s += pT[i];
  red[tid] = s; __syncthreads();
  for (int off = 128; off > 0; off >>= 1) { if (tid < off) red[tid] += red[tid + off]; __syncthreads(); }
  float lossT = red[0]; __syncthreads();
  s = 0.f;
  for (int i = tid; i < nF; i += 256) s += pF[i];
  red[tid] = s; __syncthreads();
  for (int off = 128; off > 0; off >>= 1) { if (tid < off) red[tid] += red[tid + off]; __syncthreads(); }
  if (tid == 0) out[0] = lossT / (float)(B_DIM * T_DIM * T_DIM) + red[0] / (float)(B_DIM * F_DIM * F_DIM);
}

// ---------------------------------------------------------------------------

extern "C" void kernel_launch(void* const* d_in, const int* in_sizes, int n_in,
                              void* d_out, int out_size, void* d_ws, size_t ws_size,
                              hipStream_t stream) {
  (void)in_sizes; (void)n_in; (void)out_size; (void)ws_size;
  const float* feat_s = (const float*)d_in[0];
  const float* feat_t = (const float*)d_in[1];
  // MLP params: [WqT,bqT,gqT,BqT, WkT,bkT,gkT,BkT, WqF,bqF,gqF,BqF, WkF,bkF,gkF,BkF] = d_in[2..17]
  float* ws = (float*)d_ws;

  const size_t NT = (size_t)B_DIM * T_DIM * T_DIM;   // 294912
  const size_t NF = (size_t)B_DIM * F_DIM * F_DIM;   // 131072
  size_t o = 0;
  float* GTsP = ws + o; o += NSPLIT * NT;
  float* GTtP = ws + o; o += NSPLIT * NT;
  float* GFsP = ws + o; o += NSPLIT * NF;
  float* GFtP = ws + o; o += NSPLIT * NF;
  float* GTs  = ws + o; o += NT;
  float* GTt  = ws + o; o += NT;
  float* GFs  = ws + o; o += NF;
  float* GFt  = ws + o; o += NF;
  float* nTs  = ws + o; o += B_DIM * T_DIM;
  float* nTt  = ws + o; o += B_DIM * T_DIM;
  float* nFs  = ws + o; o += B_DIM * F_DIM;
  float* nFt  = ws + o; o += B_DIM * F_DIM;
  float* rTs  = ws + o; o += B_DIM * T_DIM;
  float* rTt  = ws + o; o += B_DIM * T_DIM;
  float* rFs  = ws + o; o += B_DIM * F_DIM;
  float* rFt  = ws + o; o += B_DIM * F_DIM;
  float* qT   = ws + o; o += B_DIM * T_DIM * 4;
  float* kT   = ws + o; o += B_DIM * T_DIM * 4;
  float* qF   = ws + o; o += B_DIM * F_DIM * 4;
  float* kF   = ws + o; o += B_DIM * F_DIM * 4;
  float* aT   = ws + o; o += NT;
  float* aF   = ws + o; o += NF;
  float* eT   = ws + o; o += B_DIM * T_DIM;
  float* eF   = ws + o; o += B_DIM * F_DIM;
  float* pT   = ws + o; o += B_DIM * T_DIM;
  float* pF   = ws + o; o += B_DIM * F_DIM;

  const dim3 blk(256);
  const float inv_scale = (float)(1.0 / sqrt(4.0 + 1e-8));

  // 1) Gram matrices (WMMA f32 16x16x4), split-K partials
  gram_time_kernel<<<dim3(3, 3, B_DIM * NSPLIT), blk, 0, stream>>>(feat_s, GTsP);
  gram_time_kernel<<<dim3(3, 3, B_DIM * NSPLIT), blk, 0, stream>>>(feat_t, GTtP);
  gram_freq_kernel<<<dim3(2, 2, B_DIM * NSPLIT), blk, 0, stream>>>(feat_s, GFsP);
  gram_freq_kernel<<<dim3(2, 2, B_DIM * NSPLIT), blk, 0, stream>>>(feat_t, GFtP);

  // 2) split-K reduction
  reduce4_kernel<<<(NT + 255) / 256, blk, 0, stream>>>(GTsP, GTs, (int)NT);
  reduce4_kernel<<<(NT + 255) / 256, blk, 0, stream>>>(GTtP, GTt, (int)NT);
  reduce4_kernel<<<(NF + 255) / 256, blk, 0, stream>>>(GFsP, GFs, (int)NF);
  reduce4_kernel<<<(NF + 255) / 256, blk, 0, stream>>>(GFtP, GFt, (int)NF);

  // 3) row norms from Gram diagonal
  norms_kernel<<<(B_DIM * T_DIM + 255) / 256, blk, 0, stream>>>(GTs, nTs, T_DIM);
  norms_kernel<<<(B_DIM * T_DIM + 255) / 256, blk, 0, stream>>>(GTt, nTt, T_DIM);
  norms_kernel<<<(B_DIM * F_DIM + 255) / 256, blk, 0, stream>>>(GFs, nFs, F_DIM);
  norms_kernel<<<(B_DIM * F_DIM + 255) / 256, blk, 0, stream>>>(GFt, nFt, F_DIM);

  // 4) G -> P in place + row means
  p_rowmean_kernel<<<B_DIM * T_DIM, blk, 0, stream>>>(GTs, nTs, rTs, T_DIM);
  p_rowmean_kernel<<<B_DIM * T_DIM, blk, 0, stream>>>(GTt, nTt, rTt, T_DIM);
  p_rowmean_kernel<<<B_DIM * F_DIM, blk, 0, stream>>>(GFs, nFs, rFs, F_DIM);
  p_rowmean_kernel<<<B_DIM * F_DIM, blk, 0, stream>>>(GFt, nFt, rFt, F_DIM);

  // 5) tiny MLPs -> q/k
  mlp_kernel<<<(B_DIM * T_DIM + 255) / 256, blk, 0, stream>>>(rTs, (const float*)d_in[2],  (const float*)d_in[3],  (const float*)d_in[4],  (const float*)d_in[5],  qT, B_DIM * T_DIM);
  mlp_kernel<<<(B_DIM * T_DIM + 255) / 256, blk, 0, stream>>>(rTt, (const float*)d_in[6],  (const float*)d_in[7],  (const float*)d_in[8],  (const float*)d_in[9],  kT, B_DIM * T_DIM);
  mlp_kernel<<<(B_DIM * F_DIM + 255) / 256, blk, 0, stream>>>(rFs, (const float*)d_in[10], (const float*)d_in[11], (const float*)d_in[12], (const float*)d_in[13], qF, B_DIM * F_DIM);
  mlp_kernel<<<(B_DIM * F_DIM + 255) / 256, blk, 0, stream>>>(rFt, (const float*)d_in[14], (const float*)d_in[15], (const float*)d_in[16], (const float*)d_in[17], kF, B_DIM * F_DIM);

  // 6) attention weights
  alpha_kernel<<<B_DIM * T_DIM, blk, 0, stream>>>(qT, kT, aT, T_DIM, inv_scale);
  alpha_kernel<<<B_DIM * F_DIM, blk, 0, stream>>>(qF, kF, aF, F_DIM, inv_scale);

  // 7) row softmaxes: s-side -> log(softmax+eps); t-side -> softmax+eps and entropy
  softmax_log_kernel<<<B_DIM * T_DIM, blk, 0, stream>>>(GTs, T_DIM);
  softmax_log_kernel<<<B_DIM * F_DIM, blk, 0, stream>>>(GFs, F_DIM);
  softmax_ent_kernel<<<B_DIM * T_DIM, blk, 0, stream>>>(GTt, eT, T_DIM);
  softmax_ent_kernel<<<B_DIM * F_DIM, blk, 0, stream>>>(GFt, eF, F_DIM);

  // 8) alpha-weighted pairwise KL -> per-(b,s) partials, then deterministic finalize
  kl_loss_kernel<<<B_DIM * T_DIM, blk, 0, stream>>>(GTt, GTs, aT, eT, pT, T_DIM);
  kl_loss_kernel<<<B_DIM * F_DIM, blk, 0, stream>>>(GFt, GFs, aF, eF, pF, F_DIM);
  finalize_kernel<<<1, blk, 0, stream>>>(pT, B_DIM * T_DIM, pF, B_DIM * F_DIM, (float*)d_out);
}